// QFN_28905129902498
// MI455X (gfx1250) — compile-verified
//
#include <hip/hip_runtime.h>

typedef __attribute__((ext_vector_type(2))) float v2f;
typedef __attribute__((ext_vector_type(8))) float v8f;

#define LDS_STRIDE 17

// ---------- complex helpers ----------
__device__ __forceinline__ float2 cmul(float2 a, float2 b) {
    return make_float2(a.x * b.x - a.y * b.y, a.x * b.y + a.y * b.x);
}
__device__ __forceinline__ float2 cadd(float2 a, float2 b) {
    return make_float2(a.x + b.x, a.y + b.y);
}

// apply generic 2x2 complex gate on wire w (qubit w <-> bit (2-w), stride 4>>w)
__device__ __forceinline__ void apply_gate(float2* st, int w,
                                           float2 u00, float2 u01,
                                           float2 u10, float2 u11) {
    int stv = 4 >> w;
#pragma unroll
    for (int p = 0; p < 8; ++p) {
        if ((p & stv) == 0) {
            int q = p + stv;
            float2 a = st[p], b = st[q];
            st[p] = cadd(cmul(u00, a), cmul(u01, b));
            st[q] = cadd(cmul(u10, a), cmul(u11, b));
        }
    }
}

// RX gate in real form: U = [[c, -i*s],[-i*s, c]] applied to (uR,uI)[8] on wire w
__device__ __forceinline__ void apply_rx(float* uR, float* uI, int w, float c, float sv) {
    int stv = 4 >> w;
#pragma unroll
    for (int p = 0; p < 8; ++p) {
        if ((p & stv) == 0) {
            int q = p + stv;
            float aR = uR[p], aI = uI[p], bR = uR[q], bI = uI[q];
            uR[p] = c * aR + sv * bI;
            uI[p] = c * aI - sv * bR;
            uR[q] = sv * aI + c * bR;
            uI[q] = -sv * aR + c * bI;
        }
    }
}

// ---------------------------------------------------------------------------
// Prep kernel: build W_l (8x8 complex ansatz unitary, l=0,1) by simulating the
// ansatz on the 8 basis states, then emit real-expanded B matrices (16x16) s.t.
// Y[n,:] = X[n,:] * B with X = [Re psi(0..7), Im psi(0..7)].
// ws layout (floats): [0..255]=B0, [256..511]=B1, [512..767]=W complex pairs
// ---------------------------------------------------------------------------
__global__ void qfn_prepare(const float* __restrict__ q, float* __restrict__ ws) {
    float* Bg = ws;
    float2* Wg = (float2*)(ws + 512);
    int tid = threadIdx.x;

    if (tid < 16) {
        int l = tid >> 3, b = tid & 7;
        float2 st[8];
#pragma unroll
        for (int i = 0; i < 8; ++i) st[i] = make_float2(0.f, 0.f);
        st[b] = make_float2(1.f, 0.f);

        for (int j = 0; j < 2; ++j) {
            for (int w = 0; w < 3; ++w) {
                const float* p = q + ((l * 2 + j) * 3 + w) * 3;
                float phi = p[0], th = p[1], om = p[2];
                float c, s, ca, sa, cd, sd;
                __sincosf(0.5f * th, &s, &c);
                __sincosf(0.5f * (phi + om), &sa, &ca);
                __sincosf(0.5f * (phi - om), &sd, &cd);
                float2 u00 = make_float2(c * ca, -c * sa);   //  e^{-i(phi+om)/2} c
                float2 u01 = make_float2(-s * cd, -s * sd);  // -e^{+i(phi-om)/2} s
                float2 u10 = make_float2(s * cd, -s * sd);   //  e^{-i(phi-om)/2} s
                float2 u11 = make_float2(c * ca, c * sa);    //  e^{+i(phi+om)/2} c
                apply_gate(st, w, u00, u01, u10, u11);
            }
            int r = j + 1;  // j % (NQ-1) + 1
            for (int w = 0; w < 3; ++w) {
                int t = (w + r) % 3;
                int bc = 2 - w, bt = 2 - t;
                float2 tmp[8];
#pragma unroll
                for (int i2 = 0; i2 < 8; ++i2)
                    tmp[i2] = ((i2 >> bc) & 1) ? st[i2 ^ (1 << bt)] : st[i2];
#pragma unroll
                for (int i2 = 0; i2 < 8; ++i2) st[i2] = tmp[i2];
            }
        }
#pragma unroll
        for (int i = 0; i < 8; ++i) Wg[l * 64 + i * 8 + b] = st[i];  // W[i][b]
    }
    __syncthreads();

    // Real expansion: B[j][i]   = WR[i][j],   B[j][i+8]   = WI[i][j]
    //                 B[j+8][i] = -WI[i][j],  B[j+8][i+8] = WR[i][j]
    for (int k = 0; k < 8; ++k) {
        int idx = tid * 8 + k;  // 64 threads * 8 = 512 entries
        int l = idx >> 8, rem = idx & 255;
        int jj = rem >> 4, ii = rem & 15;
        float v;
        if (jj < 8) {
            float2 w2 = Wg[l * 64 + ((ii < 8) ? ii : (ii - 8)) * 8 + jj];
            v = (ii < 8) ? w2.x : w2.y;
        } else {
            float2 w2 = Wg[l * 64 + ((ii < 8) ? ii : (ii - 8)) * 8 + (jj - 8)];
            v = (ii < 8) ? -w2.y : w2.x;
        }
        Bg[idx] = v;
    }
}

// ---------------------------------------------------------------------------
// Main kernel: 256 items/block. psi = W2 * R(x) * W1 * R(x) * e0
// ---------------------------------------------------------------------------
__global__ void __launch_bounds__(256) qfn_main(const float* __restrict__ x,
                                                const float* __restrict__ Bg,
                                                float* __restrict__ out,
                                                int Ntot) {
    __shared__ float Xs[256 * LDS_STRIDE];
    const int tid = threadIdx.x;
    const int n = blockIdx.x * 256 + tid;
    const int lane = tid & 31;
    const int wv = tid >> 5;
    const int mrow = lane & 15;          // M (A) / N (B,C) index for this lane
    const int koff = (lane >> 4) << 1;   // K base for this lane half

    // Load both B matrices into per-lane WMMA B-layout:
    // B 4x16 chunk cc: lane holds col=mrow, rows k = cc*4 + koff + {0,1}
    v2f b1v[4], b2v[4];
#pragma unroll
    for (int cc = 0; cc < 4; ++cc) {
        int k0 = cc * 4 + koff;
        b1v[cc].x = Bg[(k0 + 0) * 16 + mrow];
        b1v[cc].y = Bg[(k0 + 1) * 16 + mrow];
        b2v[cc].x = Bg[256 + (k0 + 0) * 16 + mrow];
        b2v[cc].y = Bg[256 + (k0 + 1) * 16 + mrow];
    }

    // Per-item angles (x reused by both fusion layers)
    float c0, s0, c1, s1, c2, s2;
    __sincosf(0.5f * x[n * 3 + 0], &s0, &c0);
    __sincosf(0.5f * x[n * 3 + 1], &s1, &c1);
    __sincosf(0.5f * x[n * 3 + 2], &s2, &c2);

    // Stage 1: t = R(x)|000>
    float uR[8], uI[8];
#pragma unroll
    for (int i = 0; i < 8; ++i) { uR[i] = 0.f; uI[i] = 0.f; }
    uR[0] = 1.f;
    apply_rx(uR, uI, 0, c0, s0);
    apply_rx(uR, uI, 1, c1, s1);
    apply_rx(uR, uI, 2, c2, s2);
#pragma unroll
    for (int i = 0; i < 8; ++i) {
        Xs[tid * LDS_STRIDE + i] = uR[i];
        Xs[tid * LDS_STRIDE + 8 + i] = uI[i];
    }
    __syncthreads();

    // GEMM1: Y = X * B1, wave wv owns tiles {2wv, 2wv+1} (its own 32 rows)
#pragma unroll
    for (int tt = 0; tt < 2; ++tt) {
        int tb = (wv * 2 + tt) * 16;
        v8f acc = {};
#pragma unroll
        for (int cc = 0; cc < 4; ++cc) {
            v2f av;
            av.x = Xs[(tb + mrow) * LDS_STRIDE + cc * 4 + koff + 0];
            av.y = Xs[(tb + mrow) * LDS_STRIDE + cc * 4 + koff + 1];
            acc = __builtin_amdgcn_wmma_f32_16x16x4_f32(
                false, av, false, b1v[cc], (short)0, acc, false, false);
        }
        // C/D layout: lane holds col=mrow, row r + 8*(lane>>4) in vgpr r
#pragma unroll
        for (int r = 0; r < 8; ++r)
            Xs[(tb + r + ((lane >> 4) << 3)) * LDS_STRIDE + mrow] = acc[r];
    }
    __syncthreads();

    // Stage 2: v = R(x) u (second fusion layer's RX block)
#pragma unroll
    for (int i = 0; i < 8; ++i) {
        uR[i] = Xs[tid * LDS_STRIDE + i];
        uI[i] = Xs[tid * LDS_STRIDE + 8 + i];
    }
    apply_rx(uR, uI, 0, c0, s0);
    apply_rx(uR, uI, 1, c1, s1);
    apply_rx(uR, uI, 2, c2, s2);
    __syncthreads();
#pragma unroll
    for (int i = 0; i < 8; ++i) {
        Xs[tid * LDS_STRIDE + i] = uR[i];
        Xs[tid * LDS_STRIDE + 8 + i] = uI[i];
    }
    __syncthreads();

    // GEMM2: psi = v * B2
#pragma unroll
    for (int tt = 0; tt < 2; ++tt) {
        int tb = (wv * 2 + tt) * 16;
        v8f acc = {};
#pragma unroll
        for (int cc = 0; cc < 4; ++cc) {
            v2f av;
            av.x = Xs[(tb + mrow) * LDS_STRIDE + cc * 4 + koff + 0];
            av.y = Xs[(tb + mrow) * LDS_STRIDE + cc * 4 + koff + 1];
            acc = __builtin_amdgcn_wmma_f32_16x16x4_f32(
                false, av, false, b2v[cc], (short)0, acc, false, false);
        }
#pragma unroll
        for (int r = 0; r < 8; ++r)
            Xs[(tb + r + ((lane >> 4) << 3)) * LDS_STRIDE + mrow] = acc[r];
    }
    __syncthreads();

    // Output: stack([real, imag]) -> out[0][n][i], out[1][n][i]
    size_t imagBase = (size_t)Ntot * 8;
#pragma unroll
    for (int i = 0; i < 8; ++i) {
        out[(size_t)n * 8 + i] = Xs[tid * LDS_STRIDE + i];
        out[imagBase + (size_t)n * 8 + i] = Xs[tid * LDS_STRIDE + 8 + i];
    }
}

extern "C" void kernel_launch(void* const* d_in, const int* in_sizes, int n_in,
                              void* d_out, int out_size, void* d_ws, size_t ws_size,
                              hipStream_t stream) {
    const float* x = (const float*)d_in[0];       // (T, B, 3) f32
    const float* q = (const float*)d_in[1];       // (2, 2, 3, 3) f32
    float* out = (float*)d_out;                   // (2, T, B, 8) f32
    float* ws = (float*)d_ws;

    int Ntot = in_sizes[0] / 3;                   // T*B = 524288 (multiple of 256)

    qfn_prepare<<<1, 64, 0, stream>>>(q, ws);
    qfn_main<<<Ntot / 256, 256, 0, stream>>>(x, ws, out, Ntot);
}